// TropicalMultiHeadAttn_7524782702955
// MI455X (gfx1250) — compile-verified
//
#include <hip/hip_runtime.h>
#include <hip/hip_bf16.h>
#include <math.h>

// Tropical multi-head attention for MI455X (gfx1250).
// fp32 end-to-end (matches reference). GEMMs use V_WMMA_F32_16X16X4_F32.
// Tropical (max-plus) stages are LDS-tiled VALU with b128 LDS traffic.

typedef float v2f __attribute__((ext_vector_type(2)));
typedef float v8f __attribute__((ext_vector_type(8)));

#define H_    8
#define DH_   64
#define D_    512
#define S_    512
#define B_    2
#define MROWS (S_ * B_)             // 1024 rows of [S,B,*]
#define TENS_ELEMS (16 * 512 * 64)  // one scrambled q/k/v tensor
#define EPSV  1e-5f
#define NEG_HUGE (-3.402823466e38f)
#define POS_HUGE ( 3.402823466e38f)

// ---------------------------------------------------------------------------
// Kernel 1: fused QKV projection.  Y = h @ [Wq;Wkv]^T, then
// t = log1p(relu(y)) - lam, scattered into scrambled [bh=16, s'=512, d=64]
// layout:  bh = b*8 + s/64 ; s' = (s%64)*8 + head ; d = feat%64.
// One wave computes one 16x16 tile with WMMA f32 16x16x4, K-loop over 512.
// ---------------------------------------------------------------------------
__global__ void qkv_proj_kernel(const float* __restrict__ h,
                                const float* __restrict__ Wq,
                                const float* __restrict__ Wkv,
                                const float* __restrict__ lam,
                                float* __restrict__ qkv_ws) {
  const int lane = threadIdx.x & 31;
  const int wave = threadIdx.x >> 5;
  const int tn = blockIdx.x * 8 + wave;   // 0..95
  const int tm = blockIdx.y;              // 0..63
  const int m0 = tm * 16, n0 = tn * 16;

  const int rsel = lane & 15;             // row-in-tile this lane feeds
  const int koff = (lane >> 4) * 2;       // K sub-pair: lanes 16-31 hold K+2,K+3

  const float* xrow = h + (size_t)(m0 + rsel) * D_ + koff;
  const float* wbase = (n0 < D_) ? (Wq + (size_t)(n0 + rsel) * D_)
                                 : (Wkv + (size_t)(n0 - D_ + rsel) * D_);
  const float* wrow = wbase + koff;

  v8f c = {0.f, 0.f, 0.f, 0.f, 0.f, 0.f, 0.f, 0.f};
#pragma unroll 8
  for (int k = 0; k < D_; k += 4) {
    v2f a = *(const v2f*)(xrow + k);
    v2f b = *(const v2f*)(wrow + k);
    c = __builtin_amdgcn_wmma_f32_16x16x4_f32(false, a, false, b,
                                              (short)0, c, false, false);
  }

  // Epilogue: tropical normalize + scramble scatter.
  const int n    = n0 + rsel;          // this lane's output feature column
  const int tens = n >> 9;             // 0=q 1=k 2=v
  const int f    = n & (D_ - 1);       // feature within tensor, 0..511
  const float lamv = lam[f];
  const int hh = f >> 6;               // head
  const int d  = f & 63;
  float* base = qkv_ws + (size_t)tens * TENS_ELEMS;
#pragma unroll
  for (int i = 0; i < 8; ++i) {
    const int m = m0 + i + ((lane >> 4) << 3);   // C row: i (+8 for hi half)
    const int s = m >> 1;                        // rows are [S,B] flattened, B=2
    const int b = m & 1;
    const float t = log1pf(fmaxf(c[i], 0.0f)) - lamv;
    const int bh = b * 8 + (s >> 6);
    const int sp = ((s & 63) << 3) + hh;
    base[(((size_t)bh << 9) + sp) * 64 + d] = t;
  }
}

// ---------------------------------------------------------------------------
// Kernel 2: in-place tropical linear  x[r,o] = max_i(x[r,i] + W[o,i]).
// W staged in LDS (stride 65 -> conflict-free), row broadcast via __shfl.
// grid = (256, 3): y picks tensor/weight; each wave does 4 rows.
// ---------------------------------------------------------------------------
__global__ void trop_linear_kernel(const float* __restrict__ Wqt,
                                   const float* __restrict__ Wkt,
                                   const float* __restrict__ Wvt,
                                   float* __restrict__ qkv_ws) {
  __shared__ float Wl[64 * 65];
  const int tens = blockIdx.y;
  const float* W = (tens == 0) ? Wqt : (tens == 1) ? Wkt : Wvt;
  for (int e = threadIdx.x; e < 64 * 64; e += 256) {
    Wl[(e >> 6) * 65 + (e & 63)] = W[e];
  }
  __syncthreads();

  const int lane = threadIdx.x & 31;
  const int wave = threadIdx.x >> 5;
  float* base = qkv_ws + (size_t)tens * TENS_ELEMS;
  const int r0 = (blockIdx.x * 8 + wave) * 4;
  for (int rr = 0; rr < 4; ++rr) {
    float* row = base + (size_t)(r0 + rr) * 64;
    const float xlo = row[lane];
    const float xhi = row[lane + 32];
    float a0 = NEG_HUGE, a1 = NEG_HUGE;
#pragma unroll 8
    for (int i = 0; i < 32; ++i) {
      const float xi = __shfl(xlo, i, 32);
      a0 = fmaxf(a0, xi + Wl[lane * 65 + i]);
      a1 = fmaxf(a1, xi + Wl[(lane + 32) * 65 + i]);
    }
#pragma unroll 8
    for (int i = 0; i < 32; ++i) {
      const float xi = __shfl(xhi, i, 32);
      a0 = fmaxf(a0, xi + Wl[lane * 65 + 32 + i]);
      a1 = fmaxf(a1, xi + Wl[(lane + 32) * 65 + 32 + i]);
    }
    row[lane]      = a0;   // safe: inputs already in registers
    row[lane + 32] = a1;
  }
}

// ---------------------------------------------------------------------------
// Kernel 3: tropical attention (the roofline-dominant VALU stage).
// Workgroup = (bh, 32-row q tile).  k/v tiles double-buffered through regs,
// all LDS traffic is b128 (float4 rows, stride 17 -> 16B aligned, minimal
// banking).  Phase 1 reuses each k-chunk across the wave's 4 q rows.
// Phase 2 keeps 8 context accumulators per thread in registers.
// ---------------------------------------------------------------------------
__global__ void trop_attn_kernel(const float* __restrict__ qkv_ws,
                                 float* __restrict__ attn_vec) {
  __shared__ float4 qs[32 * 17];
  __shared__ float4 ks[32 * 17];
  __shared__ float4 vs[32 * 17];
  __shared__ float  sc[32 * 33];
  const int bh  = blockIdx.y;   // 0..15
  const int sqt = blockIdx.x;   // 0..15
  const int tid = threadIdx.x;

  const float4* qb = (const float4*)(qkv_ws + (((size_t)bh << 9) + sqt * 32) * 64);
  const float4* kb = (const float4*)(qkv_ws + (size_t)TENS_ELEMS +
                                     ((size_t)bh << 9) * 64);
  const float4* vb = kb + (TENS_ELEMS / 4);

  // q tile: 512 float4s, 2 per thread (written once, read all iterations)
  qs[(tid >> 4) * 17 + (tid & 15)]        = qb[tid];
  qs[((tid >> 4) + 16) * 17 + (tid & 15)] = qb[tid + 256];

  float ctx[8];
#pragma unroll
  for (int j = 0; j < 8; ++j) ctx[j] = NEG_HUGE;

  const int sq1 = tid >> 5;     // phase1: all lanes of a wave share sq -> bcast
  const int sk1 = tid & 31;     // phase1: sk = lane
  const int sq2 = tid >> 3;     // phase2 owner row
  const int dg  = tid & 7;      // phase2 d-octet

  // software pipeline: preload tile 0 into registers
  float4 kr0 = kb[tid], kr1 = kb[tid + 256];
  float4 vr0 = vb[tid], vr1 = vb[tid + 256];

  for (int kt = 0; kt < 16; ++kt) {
    __syncthreads();            // previous iteration's sc/vs readers done
    ks[(tid >> 4) * 17 + (tid & 15)]        = kr0;
    ks[((tid >> 4) + 16) * 17 + (tid & 15)] = kr1;
    vs[(tid >> 4) * 17 + (tid & 15)]        = vr0;
    vs[((tid >> 4) + 16) * 17 + (tid & 15)] = vr1;
    __syncthreads();
    if (kt < 15) {              // issue next tile's global loads early
      const int nb = (kt + 1) * 512;
      kr0 = kb[nb + tid]; kr1 = kb[nb + tid + 256];
      vr0 = vb[nb + tid]; vr1 = vb[nb + tid + 256];
    }

    // ---- phase 1: score tile ----
    float mx0 = NEG_HUGE, mx1 = NEG_HUGE, mx2 = NEG_HUGE, mx3 = NEG_HUGE;
    float mn0 = POS_HUGE, mn1 = POS_HUGE, mn2 = POS_HUGE, mn3 = POS_HUGE;
#pragma unroll 4
    for (int g = 0; g < 16; ++g) {
      const float4 kv = ks[sk1 * 17 + g];
      {
        const float4 qv = qs[sq1 * 17 + g];
        const float d0 = qv.x - kv.x, d1 = qv.y - kv.y;
        const float d2 = qv.z - kv.z, d3 = qv.w - kv.w;
        mx0 = fmaxf(mx0, fmaxf(fmaxf(d0, d1), fmaxf(d2, d3)));
        mn0 = fminf(mn0, fminf(fminf(d0, d1), fminf(d2, d3)));
      }
      {
        const float4 qv = qs[(sq1 + 8) * 17 + g];
        const float d0 = qv.x - kv.x, d1 = qv.y - kv.y;
        const float d2 = qv.z - kv.z, d3 = qv.w - kv.w;
        mx1 = fmaxf(mx1, fmaxf(fmaxf(d0, d1), fmaxf(d2, d3)));
        mn1 = fminf(mn1, fminf(fminf(d0, d1), fminf(d2, d3)));
      }
      {
        const float4 qv = qs[(sq1 + 16) * 17 + g];
        const float d0 = qv.x - kv.x, d1 = qv.y - kv.y;
        const float d2 = qv.z - kv.z, d3 = qv.w - kv.w;
        mx2 = fmaxf(mx2, fmaxf(fmaxf(d0, d1), fmaxf(d2, d3)));
        mn2 = fminf(mn2, fminf(fminf(d0, d1), fminf(d2, d3)));
      }
      {
        const float4 qv = qs[(sq1 + 24) * 17 + g];
        const float d0 = qv.x - kv.x, d1 = qv.y - kv.y;
        const float d2 = qv.z - kv.z, d3 = qv.w - kv.w;
        mx3 = fmaxf(mx3, fmaxf(fmaxf(d0, d1), fmaxf(d2, d3)));
        mn3 = fminf(mn3, fminf(fminf(d0, d1), fminf(d2, d3)));
      }
    }
    sc[sq1 * 33 + sk1]        = mn0 - mx0;   // == -(max - min)
    sc[(sq1 + 8) * 33 + sk1]  = mn1 - mx1;
    sc[(sq1 + 16) * 33 + sk1] = mn2 - mx2;
    sc[(sq1 + 24) * 33 + sk1] = mn3 - mx3;
    __syncthreads();

    // ---- phase 2: max-plus aggregation ----
#pragma unroll 4
    for (int sk = 0; sk < 32; ++sk) {
      const float s = sc[sq2 * 33 + sk];
      const float4 v0 = vs[sk * 17 + dg * 2];
      const float4 v1 = vs[sk * 17 + dg * 2 + 1];
      ctx[0] = fmaxf(ctx[0], s + v0.x);
      ctx[1] = fmaxf(ctx[1], s + v0.y);
      ctx[2] = fmaxf(ctx[2], s + v0.z);
      ctx[3] = fmaxf(ctx[3], s + v0.w);
      ctx[4] = fmaxf(ctx[4], s + v1.x);
      ctx[5] = fmaxf(ctx[5], s + v1.y);
      ctx[6] = fmaxf(ctx[6], s + v1.z);
      ctx[7] = fmaxf(ctx[7], s + v1.w);
    }
  }

  // expm1 + inverse scramble:  attn_vec[s2, b2, h2*64+d]
  const int b2 = bh >> 3, h2 = bh & 7;
  const int s2 = sqt * 32 + sq2;
  float4* out = (float4*)(attn_vec + ((size_t)s2 * B_ + b2) * (H_ * DH_) +
                          h2 * 64 + dg * 8);
  float4 o0, o1;
  o0.x = expm1f(ctx[0]); o0.y = expm1f(ctx[1]);
  o0.z = expm1f(ctx[2]); o0.w = expm1f(ctx[3]);
  o1.x = expm1f(ctx[4]); o1.y = expm1f(ctx[5]);
  o1.z = expm1f(ctx[6]); o1.w = expm1f(ctx[7]);
  out[0] = o0;
  out[1] = o1;
}

// ---------------------------------------------------------------------------
// Kernel 4: output projection  y = attn_vec @ Wo^T + h   (WMMA f32 16x16x4)
// ---------------------------------------------------------------------------
__global__ void out_proj_kernel(const float* __restrict__ attn_vec,
                                const float* __restrict__ Wo,
                                const float* __restrict__ h,
                                float* __restrict__ yws) {
  const int lane = threadIdx.x & 31;
  const int wave = threadIdx.x >> 5;
  const int tn = blockIdx.x * 8 + wave;  // 0..31
  const int tm = blockIdx.y;             // 0..63
  const int m0 = tm * 16, n0 = tn * 16;
  const int rsel = lane & 15;
  const int koff = (lane >> 4) * 2;

  const float* xrow = attn_vec + (size_t)(m0 + rsel) * D_ + koff;
  const float* wrow = Wo + (size_t)(n0 + rsel) * D_ + koff;

  v8f c = {0.f, 0.f, 0.f, 0.f, 0.f, 0.f, 0.f, 0.f};
#pragma unroll 8
  for (int k = 0; k < D_; k += 4) {
    v2f a = *(const v2f*)(xrow + k);
    v2f b = *(const v2f*)(wrow + k);
    c = __builtin_amdgcn_wmma_f32_16x16x4_f32(false, a, false, b,
                                              (short)0, c, false, false);
  }
  const int n = n0 + rsel;
#pragma unroll
  for (int i = 0; i < 8; ++i) {
    const int m = m0 + i + ((lane >> 4) << 3);
    yws[(size_t)m * D_ + n] = c[i] + h[(size_t)m * D_ + n];
  }
}

// ---------------------------------------------------------------------------
// Kernel 5: LayerNorm over rows of 512.  One 256-thread block per row.
// ---------------------------------------------------------------------------
__global__ void layernorm_kernel(const float* __restrict__ yws,
                                 const float* __restrict__ gamma,
                                 const float* __restrict__ beta,
                                 float* __restrict__ out) {
  __shared__ float rs1[256];
  __shared__ float rs2[256];
  const int m = blockIdx.x;
  const int t = threadIdx.x;
  const float e0 = yws[(size_t)m * D_ + t];
  const float e1 = yws[(size_t)m * D_ + t + 256];
  rs1[t] = e0 + e1;
  rs2[t] = e0 * e0 + e1 * e1;
  __syncthreads();
  for (int off = 128; off > 0; off >>= 1) {
    if (t < off) { rs1[t] += rs1[t + off]; rs2[t] += rs2[t + off]; }
    __syncthreads();
  }
  const float mu  = rs1[0] * (1.0f / 512.0f);
  const float var = rs2[0] * (1.0f / 512.0f) - mu * mu;
  const float rstd = rsqrtf(var + EPSV);
  out[(size_t)m * D_ + t]       = (e0 - mu) * rstd * gamma[t]       + beta[t];
  out[(size_t)m * D_ + t + 256] = (e1 - mu) * rstd * gamma[t + 256] + beta[t + 256];
}

// ---------------------------------------------------------------------------
extern "C" void kernel_launch(void* const* d_in, const int* in_sizes, int n_in,
                              void* d_out, int out_size, void* d_ws, size_t ws_size,
                              hipStream_t stream) {
  const float* h     = (const float*)d_in[0];
  const float* Wq    = (const float*)d_in[1];
  const float* Wkv   = (const float*)d_in[2];
  const float* Wqt   = (const float*)d_in[3];
  const float* Wkt   = (const float*)d_in[4];
  const float* Wvt   = (const float*)d_in[5];
  const float* lam   = (const float*)d_in[6];
  const float* Wo    = (const float*)d_in[7];
  const float* gamma = (const float*)d_in[8];
  const float* beta  = (const float*)d_in[9];
  float* out = (float*)d_out;

  // Workspace layout (floats): [q|k|v] scrambled tensors (3 x 512K), then
  // attn_vec (512K).  y (=attn_out + h) reuses the q region after attention.
  float* ws  = (float*)d_ws;                 // needs 4 * 2 MB = 8 MB
  float* qkv = ws;
  float* av  = ws + (size_t)3 * TENS_ELEMS;
  float* yws = ws;                           // q region is dead by then

  qkv_proj_kernel   <<<dim3(12, 64), 256, 0, stream>>>(h, Wq, Wkv, lam, qkv);
  trop_linear_kernel<<<dim3(256, 3), 256, 0, stream>>>(Wqt, Wkt, Wvt, qkv);
  trop_attn_kernel  <<<dim3(16, 16), 256, 0, stream>>>(qkv, av);
  out_proj_kernel   <<<dim3(4, 64),  256, 0, stream>>>(av, Wo, h, yws);
  layernorm_kernel  <<<dim3(MROWS),  256, 0, stream>>>(yws, gamma, beta, out);
}